// CrossAttention_24756191494566
// MI455X (gfx1250) — compile-verified
//
#include <hip/hip_runtime.h>

typedef __attribute__((ext_vector_type(2))) float v2f;
typedef __attribute__((ext_vector_type(8))) float v8f;
typedef __attribute__((ext_vector_type(4))) unsigned int u32x4;
typedef __attribute__((ext_vector_type(8))) int i32x8;
typedef __attribute__((ext_vector_type(4))) int i32x4;

// D = A(16x4 f32) * B(4x16 f32) + C(16x16 f32), wave32
#define WMMA_F32(a, b, c) \
  __builtin_amdgcn_wmma_f32_16x16x4_f32(false, (a), false, (b), (short)0, (c), false, false)

static constexpr int B_  = 8;
static constexpr int E_  = 512;
static constexpr int N_  = 1024;   // Hs*Ws
static constexpr int L_  = 1024;
static constexpr int H_  = 8;
static constexpr int Dh_ = 64;
static constexpr int K_  = 512;          // GEMM inner dim (always E)
static constexpr long PLANE = (long)N_ * E_;  // per-batch plane (all operands)

static constexpr int KS = 32;            // K-slab staged in LDS per step
static constexpr int NSLAB = K_ / KS;    // 16
static constexpr int WROW = KS + 2;      // padded LDS row stride (bank-conflict free)

#if __has_builtin(__builtin_amdgcn_tensor_load_to_lds)
#define HAVE_TDM 1
#endif

#ifdef HAVE_TDM
// TDM: DMA a 64(e) x 32(k) fp32 tile of W (row stride 512) into LDS, inserting
// 2 DWORDs of padding every 32 DWORDs (-> LDS row stride 34 floats).
// Descriptor packing per CDNA5 ISA.
__device__ __forceinline__ void tdm_load_w_slab(const float* W, int e0, int k0,
                                                void* ldsDst) {
  unsigned long long ga = (unsigned long long)(const void*)(W + (size_t)e0 * K_ + k0);
  unsigned lds = (unsigned)(unsigned long long)ldsDst;  // generic addr low 32 = LDS offset
  u32x4 g0;
  g0[0] = 1u;                                            // count=1 (valid descriptor)
  g0[1] = lds;                                           // lds_addr
  g0[2] = (unsigned)ga;                                  // global_addr[31:0]
  g0[3] = (unsigned)((ga >> 32) & 0x1FFFFFFull) | (2u << 30);  // ga[56:32] | type=2
  // group1: data_size=4B(2), pad_enable, pad_interval=4 (32 dw), pad_amount=1 (2 dw)
  int w0 = (2 << 16) | (1 << 20) | (4 << 22) | (1 << 25);
  unsigned td0 = 512u, td1 = 512u;                       // tensor dims (in-bounds tile)
  int w1 = (int)((td0 & 0xFFFFu) << 16);                 // tensor_dim0[15:0]
  int w2 = (int)((td0 >> 16) | ((td1 & 0xFFFFu) << 16)); // tensor_dim0[31:16]|dim1[15:0]
  int w3 = (int)((td1 >> 16) | ((unsigned)KS << 16));    // dim1[31:16] | tile_dim0=32
  int w4 = 64;                                           // tile_dim1=64, tile_dim2=0
  int w5 = K_;                                           // tensor_dim0_stride = 512
  int w6 = 0, w7 = 0;
  i32x8 g1 = {w0, w1, w2, w3, w4, w5, w6, w7};
  i32x4 z4 = {0, 0, 0, 0};
#if __clang_major__ >= 23
  i32x8 z8 = {0, 0, 0, 0, 0, 0, 0, 0};
  __builtin_amdgcn_tensor_load_to_lds(g0, g1, z4, z4, z8, 0);
#else
  __builtin_amdgcn_tensor_load_to_lds(g0, g1, z4, z4, 0);
#endif
}
#endif

// ---------------------------------------------------------------------------
// WMMA f32 GEMM (compile-time strides):
//   Out[b, r, e] = sum_k A[b, r*AROW + k*AK] * W[e, k] + bias[e]
//   stored at out[b*PLANE + r*OROW + e*OE]
// Block = 256 threads = 8 waves; WG tile = 128 r x 64 e; wave tile = 16 r x 64 e.
// W slab (64e x 32k) double-buffered in LDS via TDM; shared by all 8 waves.
// ---------------------------------------------------------------------------
template <int AROW, int AK, int OROW, int OE>
__global__ __launch_bounds__(256) void gemm_wmma_kernel(
    const float* __restrict__ A, const float* __restrict__ W,
    const float* __restrict__ bias, float* __restrict__ out)
{
  __shared__ float Wlds[2][64][WROW];

  const int wave   = threadIdx.x >> 5;
  const int lane   = threadIdx.x & 31;
  const int m      = lane & 15;
  const int kq     = (lane >> 4) << 1;  // 0 | 2 per half-wave
  const int rowAdd = (lane >> 4) << 3;  // C layout: rows t / t+8
  const int r0 = blockIdx.x * 128 + wave * 16;
  const int e0 = blockIdx.y * 64;
  const int b  = blockIdx.z;

  const float* Ab = A + (long)b * PLANE;
  v8f acc0 = {}, acc1 = {}, acc2 = {}, acc3 = {};

#ifdef HAVE_TDM
  const bool tdmWave = (threadIdx.x < 32);
  if (tdmWave) tdm_load_w_slab(W, e0, 0, &Wlds[0][0][0]);
#endif

  for (int s = 0; s < NSLAB; ++s) {
    const int cur = s & 1;
#ifdef HAVE_TDM
    if (tdmWave) {
      if (s + 1 < NSLAB) {
        tdm_load_w_slab(W, e0, (s + 1) * KS, &Wlds[(s + 1) & 1][0][0]);
        __builtin_amdgcn_s_wait_tensorcnt(1);  // slab s complete (in-order TDM)
      } else {
        __builtin_amdgcn_s_wait_tensorcnt(0);
      }
    }
#else
    // Cooperative staging fallback: 256 threads x 8 floats = 64x32 tile
    {
      const int er = threadIdx.x >> 2;           // 0..63
      const int kc = (threadIdx.x & 3) << 3;     // 0,8,16,24
      const float* src = W + (size_t)(e0 + er) * K_ + s * KS + kc;
      #pragma unroll
      for (int j = 0; j < 8; ++j) Wlds[cur][er][kc + j] = src[j];
    }
#endif
    __syncthreads();

    #pragma unroll
    for (int kk = 0; kk < KS; kk += 4) {
      const int kg = s * KS + kk;
      v2f a;
      a.x = Ab[(long)(r0 + m) * AROW + (long)(kg + kq) * AK];
      a.y = Ab[(long)(r0 + m) * AROW + (long)(kg + kq + 1) * AK];
      v2f b0 = *(const v2f*)&Wlds[cur][m +  0][kk + kq];
      v2f b1 = *(const v2f*)&Wlds[cur][m + 16][kk + kq];
      v2f b2 = *(const v2f*)&Wlds[cur][m + 32][kk + kq];
      v2f b3 = *(const v2f*)&Wlds[cur][m + 48][kk + kq];
      acc0 = WMMA_F32(a, b0, acc0);
      acc1 = WMMA_F32(a, b1, acc1);
      acc2 = WMMA_F32(a, b2, acc2);
      acc3 = WMMA_F32(a, b3, acc3);
    }
    __syncthreads();
  }

  float* Ob = out + (long)b * PLANE;
  #pragma unroll
  for (int t = 0; t < 8; ++t) {
    const int r = r0 + t + rowAdd;
    const int e = e0 + m;
    Ob[(long)r * OROW + (long)(e +  0) * OE] = acc0[t] + bias[e +  0];
    Ob[(long)r * OROW + (long)(e + 16) * OE] = acc1[t] + bias[e + 16];
    Ob[(long)r * OROW + (long)(e + 32) * OE] = acc2[t] + bias[e + 32];
    Ob[(long)r * OROW + (long)(e + 48) * OE] = acc3[t] + bias[e + 48];
  }
}

// ---------------------------------------------------------------------------
// Flash attention: one WG per (128 query rows, head h, batch b).
// Qp/Kp/Vp are (B, rows, E) row-major; head slice = columns [h*64, h*64+64).
// Each wave: 16 query rows, O accum 16x64 (4 v8f), online softmax.
// ---------------------------------------------------------------------------
__global__ __launch_bounds__(256) void attn_kernel(
    const float* __restrict__ Qp, const float* __restrict__ Kp,
    const float* __restrict__ Vp, float* __restrict__ Attn)
{
  __shared__ float Plds[8][16][17];  // per-wave P tile, padded (conflict-free)
  const int wave   = threadIdx.x >> 5;
  const int lane   = threadIdx.x & 31;
  const int m      = lane & 15;
  const int kq     = (lane >> 4) << 1;
  const int rowAdd = (lane >> 4) << 3;
  const int n0 = blockIdx.x * 128 + wave * 16;
  const int h  = blockIdx.y;
  const int b  = blockIdx.z;

  const float* Qb = Qp + (long)b * N_ * E_ + h * Dh_;
  const float* Kb = Kp + (long)b * L_ * E_ + h * Dh_;
  const float* Vb = Vp + (long)b * L_ * E_ + h * Dh_;

  // Preload Q A-fragments for all 16 K-steps over Dh=64 (reused for every L tile)
  v2f qa[16];
  #pragma unroll
  for (int kk = 0; kk < 16; ++kk)
    qa[kk] = *(const v2f*)(Qb + (long)(n0 + m) * E_ + kk * 4 + kq);

  v8f o0 = {}, o1 = {}, o2 = {}, o3 = {};
  float mrow[8], lrow[8];
  #pragma unroll
  for (int t = 0; t < 8; ++t) { mrow[t] = -1e30f; lrow[t] = 0.0f; }
  const float scale = 0.125f;  // 1/sqrt(Dh)

  for (int l0 = 0; l0 < L_; l0 += 16) {
    // S = Q * K^T for this 16x16 tile (B frag: lane m is column l0+m)
    v8f s = {};
    #pragma unroll
    for (int kk = 0; kk < 16; ++kk) {
      v2f bf = *(const v2f*)(Kb + (long)(l0 + m) * E_ + kk * 4 + kq);
      s = WMMA_F32(qa[kk], bf, s);
    }
    // Online softmax per row; each 16-lane half owns rows t(+8)
    #pragma unroll
    for (int t = 0; t < 8; ++t) {
      float sv = s[t] * scale;
      float rmax = sv;
      rmax = fmaxf(rmax, __shfl_xor(rmax, 1, 32));
      rmax = fmaxf(rmax, __shfl_xor(rmax, 2, 32));
      rmax = fmaxf(rmax, __shfl_xor(rmax, 4, 32));
      rmax = fmaxf(rmax, __shfl_xor(rmax, 8, 32));
      float newm = fmaxf(mrow[t], rmax);
      float corr = __expf(mrow[t] - newm);
      float p    = __expf(sv - newm);
      float psum = p;
      psum += __shfl_xor(psum, 1, 32);
      psum += __shfl_xor(psum, 2, 32);
      psum += __shfl_xor(psum, 4, 32);
      psum += __shfl_xor(psum, 8, 32);
      lrow[t] = lrow[t] * corr + psum;
      mrow[t] = newm;
      o0[t] *= corr; o1[t] *= corr; o2[t] *= corr; o3[t] *= corr;
      Plds[wave][t + rowAdd][m] = p;  // C layout -> LDS row-major
    }
    __syncthreads();  // uniform trip count across all waves
    // O += P(16x16) * V(16x64): 4 K-steps, 4 d-subtiles
    #pragma unroll
    for (int kk = 0; kk < 4; ++kk) {
      v2f a;
      a.x = Plds[wave][m][kk * 4 + kq];
      a.y = Plds[wave][m][kk * 4 + kq + 1];
      const float* V0 = Vb + (long)(l0 + kk * 4 + kq) * E_;
      const float* V1 = V0 + E_;
      v2f vb0, vb1, vb2, vb3;
      vb0.x = V0[m +  0]; vb0.y = V1[m +  0];
      vb1.x = V0[m + 16]; vb1.y = V1[m + 16];
      vb2.x = V0[m + 32]; vb2.y = V1[m + 32];
      vb3.x = V0[m + 48]; vb3.y = V1[m + 48];
      o0 = WMMA_F32(a, vb0, o0);
      o1 = WMMA_F32(a, vb1, o1);
      o2 = WMMA_F32(a, vb2, o2);
      o3 = WMMA_F32(a, vb3, o3);
    }
    __syncthreads();
  }

  float* Ob = Attn + (long)b * N_ * E_ + h * Dh_;
  #pragma unroll
  for (int t = 0; t < 8; ++t) {
    const float inv = 1.0f / lrow[t];
    const long base = (long)(n0 + t + rowAdd) * E_;
    Ob[base + m +  0] = o0[t] * inv;
    Ob[base + m + 16] = o1[t] * inv;
    Ob[base + m + 32] = o2[t] * inv;
    Ob[base + m + 48] = o3[t] * inv;
  }
}

// ---------------------------------------------------------------------------
// Channel-wise RMSNorm over (B, E, N) in place; one thread per pixel.
// ---------------------------------------------------------------------------
__global__ __launch_bounds__(256) void rmsnorm_kernel(float* __restrict__ out,
                                                      const float* __restrict__ g)
{
  const int p = blockIdx.x * blockDim.x + threadIdx.x;  // 0 .. B*N-1
  const int b = p >> 10;
  const int n = p & (N_ - 1);
  float* base = out + (long)b * E_ * N_ + n;
  float ss = 0.0f;
  for (int c = 0; c < E_; ++c) {
    const float v = base[(long)c * N_];
    ss += v * v;
  }
  const float rinv = rsqrtf(ss * (1.0f / E_) + 1e-6f);
  for (int c = 0; c < E_; ++c)
    base[(long)c * N_] = base[(long)c * N_] * rinv * g[c];
}

// ---------------------------------------------------------------------------
extern "C" void kernel_launch(void* const* d_in, const int* in_sizes, int n_in,
                              void* d_out, int out_size, void* d_ws, size_t ws_size,
                              hipStream_t stream) {
  const float* query = (const float*)d_in[0];   // (B, E, N)
  const float* key   = (const float*)d_in[1];   // (B, L, E)
  const float* value = (const float*)d_in[2];   // (B, L, E)
  const float* Wq = (const float*)d_in[3];
  const float* bq = (const float*)d_in[4];
  const float* Wk = (const float*)d_in[5];
  const float* bk = (const float*)d_in[6];
  const float* Wv = (const float*)d_in[7];
  const float* bv = (const float*)d_in[8];
  const float* Wo = (const float*)d_in[9];
  const float* bo = (const float*)d_in[10];
  const float* g  = (const float*)d_in[11];
  float* out = (float*)d_out;

  // Workspace: 4 fp32 intermediates of (B, 1024, E) = 16 MB each (64 MB total)
  float* Qp   = (float*)d_ws;
  float* Kp   = Qp + PLANE * B_;
  float* Vp   = Kp + PLANE * B_;
  float* Attn = Vp + PLANE * B_;

  dim3 blk(256);
  dim3 gGemm(N_ / 128, E_ / 64, B_);  // (8, 8, 8)

  // Q proj: A[r=n,k=i] = query[b, i, n] (AROW=1, AK=N); Qp is (B, N, E)
  gemm_wmma_kernel<1, N_, E_, 1><<<gGemm, blk, 0, stream>>>(query, Wq, bq, Qp);
  // K proj: A[r=l,k=i] = key[b, l, i]   (AROW=E, AK=1); Kp is (B, L, E)
  gemm_wmma_kernel<E_, 1, E_, 1><<<gGemm, blk, 0, stream>>>(key, Wk, bk, Kp);
  // V proj
  gemm_wmma_kernel<E_, 1, E_, 1><<<gGemm, blk, 0, stream>>>(value, Wv, bv, Vp);

  // Attention: Attn (B, N, E) = concat over heads of softmax(QK^T/8) V
  dim3 gAttn(N_ / 128, H_, B_);  // (8, 8, 8)
  attn_kernel<<<gAttn, blk, 0, stream>>>(Qp, Kp, Vp, Attn);

  // Out proj: A[r=n,k=i] = Attn[b, n, i]; store channel-major out[b, e, n]
  gemm_wmma_kernel<E_, 1, 1, N_><<<gGemm, blk, 0, stream>>>(Attn, Wo, bo, out);

  // RMSNorm over channels, in place on d_out
  rmsnorm_kernel<<<dim3((B_ * N_) / 256), blk, 0, stream>>>(out, g);
}